// ConeBackProjection3D_28192165331443
// MI455X (gfx1250) — compile-verified
//
#include <hip/hip_runtime.h>
#include <cstdint>

// Cone-beam backprojection (FDK) for MI455X / gfx1250.
//
// Geometry on the fp32 matrix unit: V_WMMA_F32_16X16X4_F32 with
//   A = 16 projections x 4  (P-matrix row, per chunk, from LDS)
//   B = 4 x 16 voxels       (homogeneous coords, loop-invariant)
// -> D[m=proj][n=voxel]; one WMMA triple (u/v/t rows) covers
// 16 projections x 16 voxels = 3x1024 MACs, exactly the scalar FMA count.
// Lane n gathers adjacent detector pixels of the SAME slice as its
// neighbors (2 active slices per wave), all gathers are 32-bit-offset
// saddr-form. Trajectory table staged to LDS via TDM (tensor_load_to_lds,
// TENSORcnt) + async global->LDS copies (ASYNCcnt).

#define N_PROJ 360
#define DET_H  512
#define DET_W  512
#define VZ 128
#define VY 128
#define VX 128
#define PCHUNK 16
#define TDM_ELEMS (180 * 12)   // first 180 projection matrices via TDM

typedef __attribute__((ext_vector_type(2))) float    v2f;
typedef __attribute__((ext_vector_type(8))) float    v8f;
typedef __attribute__((ext_vector_type(4))) unsigned v4u;
typedef __attribute__((ext_vector_type(8))) int      v8i;
typedef __attribute__((ext_vector_type(4))) int      v4i;

#if defined(__has_builtin)
#if __has_builtin(__builtin_amdgcn_tensor_load_to_lds)
#define HAVE_TDM 1
#endif
#endif

__global__ __launch_bounds__(256)
void cone_bp_wmma_kernel(const float* __restrict__ sino,      // [1,360,512,512]
                         const float* __restrict__ vorigin,   // (oz,oy,ox)
                         const float* __restrict__ vspacing,  // (sz,sy,sx)
                         const float* __restrict__ traj,      // [360,3,4]
                         const float* __restrict__ pmul_p,    // scalar
                         float* __restrict__ out)             // [1,128,128,128]
{
    __shared__ __align__(16) float sP[N_PROJ * 12];

    const int tid  = threadIdx.x;     // 0..255 (8 waves of 32)
    const int lane = tid & 31;
    const int wave = tid >> 5;
    const int y  = blockIdx.x;
    const int z  = blockIdx.y;
    const int x0 = wave * 16;         // this wave's 16 voxels along X

    // ---- Stage trajectory matrices (17.3 KB) into LDS.
    // First half: one TDM descriptor issued by wave 0 (TENSORcnt).
    // Second half: async global->LDS per-lane copies (ASYNCcnt). ----
#ifdef HAVE_TDM
    if (wave == 0) {
        const unsigned lds_base = (unsigned)(uintptr_t)(&sP[0]);
        const unsigned long long ga = (unsigned long long)(uintptr_t)traj;
        const unsigned ne = TDM_ELEMS;
        v4u g0 = { 0x1u,                                   // count=1 valid user desc
                   lds_base,                               // lds_addr
                   (unsigned)(ga & 0xffffffffu),           // global_addr[31:0]
                   (unsigned)((ga >> 32) & 0x01ffffffu) | (2u << 30) }; // ga[56:32], type=2
        v8i g1 = { (int)0x00020000u,                       // mask=0, data_size=4B
                   (int)((ne & 0xffffu) << 16),            // tensor_dim0[15:0]
                   (int)(((ne >> 16) & 0xffffu) | (1u << 16)), // dim0 hi, tensor_dim1=1
                   (int)((ne & 0xffffu) << 16),            // tile_dim0 = ne
                   1,                                      // tile_dim1 = 1
                   (int)ne,                                // tensor_dim0_stride
                   0, 0 };
        v4i zero4 = { 0, 0, 0, 0 };
        v8i zero8 = { 0, 0, 0, 0, 0, 0, 0, 0 };
        __builtin_amdgcn_tensor_load_to_lds(g0, g1, zero4, zero4, zero8, 0);
        __builtin_amdgcn_s_wait_tensorcnt(0);
    }
    const int astart = TDM_ELEMS;
#else
    const int astart = 0;
#endif
    {
        const float* tbase = traj;
        for (int i = astart + tid; i < N_PROJ * 12; i += 256) {
            unsigned lds_off = (unsigned)(uintptr_t)(&sP[i]);
            unsigned goff    = (unsigned)(i * 4);
            asm volatile("global_load_async_to_lds_b32 %0, %1, %2"
                         :: "v"(lds_off), "v"(goff), "s"(tbase)
                         : "memory");
        }
        asm volatile("s_wait_asynccnt 0" ::: "memory");
    }
    __syncthreads();

    const float oz = vorigin[0],  oy = vorigin[1],  ox = vorigin[2];
    const float sz = vspacing[0], sy = vspacing[1], sx = vspacing[2];
    const float pm = pmul_p[0];

    const float wyv = oy + sy * (float)y;
    const float wzv = oz + sz * (float)z;
    const float wxl = ox + sx * (float)(x0 + (lane & 15));

    // B-matrix (4x16 fp32), loop-invariant: column n = voxel x0+n,
    // rows = (wx, wy, wz, 1). ISA layout: lanes 0-15 hold K=0,1; 16-31 K=2,3.
    v2f bmat;
    bmat.x = (lane < 16) ? wxl : wzv;
    bmat.y = (lane < 16) ? wyv : 1.0f;

    float acc = 0.0f;                       // this lane's voxel accumulator

    const int  nrow = lane & 15;            // A row / voxel column index
    const int  kofs = (lane >> 4) * 2;      // A-matrix K pair held by this lane
    const int  pgsel = (lane >> 4) * 8;     // projection row group in D
    const unsigned slice_elems = DET_H * DET_W;

    for (int p0 = 0; p0 < N_PROJ; p0 += PCHUNK) {
        // A-matrices: row m = projection p0+m, entries P[p][row][k] from LDS.
        const int pa  = p0 + nrow;
        const int pca = (pa < N_PROJ) ? pa : (N_PROJ - 1);
        const float* Pm = &sP[pca * 12];
        v2f au = { Pm[0 + kofs], Pm[1 + kofs] };
        v2f av = { Pm[4 + kofs], Pm[5 + kofs] };
        v2f at = { Pm[8 + kofs], Pm[9 + kofs] };

        // Prefetch next chunk's slices (emits global_prefetch_b8).
        if (pa + PCHUNK < N_PROJ)
            __builtin_prefetch(sino + (size_t)(pa + PCHUNK) * slice_elems, 0, 3);

        v8f czero = {};
        // D[m][n]: un/vn/t for projection p0+m, voxel x0+n (uniform flow, EXEC=all-1s)
        v8f du = __builtin_amdgcn_wmma_f32_16x16x4_f32(false, au, false, bmat,
                                                       (short)0, czero, false, false);
        v8f dv = __builtin_amdgcn_wmma_f32_16x16x4_f32(false, av, false, bmat,
                                                       (short)0, czero, false, false);
        v8f dt = __builtin_amdgcn_wmma_f32_16x16x4_f32(false, at, false, bmat,
                                                       (short)0, czero, false, false);

        #pragma unroll
        for (int r = 0; r < 8; ++r) {       // 8 projections for this half-wave
            const int pr  = p0 + pgsel + r;
            const int pcr = (pr < N_PROJ) ? pr : (N_PROJ - 1);
            const unsigned sbase = (unsigned)pcr * slice_elems; // 32-bit element index

            float t  = dt[r];
            float rt = __builtin_amdgcn_rcpf(t);
            float u  = du[r] * rt;
            float v  = dv[r] * rt;
            float uf = floorf(u), vf = floorf(v);
            float fu = u - uf,    fv = v - vf;
            int   u0 = (int)uf,   v0 = (int)vf;
            bool valid = (pr < N_PROJ) &&
                         (u >= 0.0f) && (u <= (float)(DET_W - 1)) &&
                         (v >= 0.0f) && (v <= (float)(DET_H - 1));
            int u0c = min(max(u0, 0),     DET_W - 1);
            int u1c = min(max(u0 + 1, 0), DET_W - 1);
            int v0c = min(max(v0, 0),     DET_H - 1);
            int v1c = min(max(v0 + 1, 0), DET_H - 1);
            // saddr-form gathers: uniform base + 32-bit per-lane offset
            unsigned row0 = sbase + ((unsigned)v0c << 9);
            unsigned row1 = sbase + ((unsigned)v1c << 9);
            float s00 = sino[row0 + (unsigned)u0c];
            float s01 = sino[row0 + (unsigned)u1c];
            float s10 = sino[row1 + (unsigned)u0c];
            float s11 = sino[row1 + (unsigned)u1c];
            float top = s00 + (s01 - s00) * fu;
            float bot = s10 + (s11 - s10) * fu;
            float bil = top + (bot - top) * fv;
            float w   = pm * rt * rt;
            acc += valid ? bil * w : 0.0f;
        }
    }

    // Pair the two projection-halves of each voxel, then lanes 0-15 do one
    // fully coalesced 16-float store per wave.
    acc += __shfl_xor(acc, 16, 32);
    if (lane < 16) {
        const size_t base = (((size_t)z * VY) + (size_t)y) * VX + (size_t)(x0 + nrow);
        out[base] = acc;
    }
}

extern "C" void kernel_launch(void* const* d_in, const int* in_sizes, int n_in,
                              void* d_out, int out_size, void* d_ws, size_t ws_size,
                              hipStream_t stream) {
    // setup_inputs() order:
    // 0: input sinogram f32 [1,360,512,512]
    // 1: volume_shape   i64 [3]   (compile-time constants; unused)
    // 2: volume_origin  f32 [3]   (z,y,x)
    // 3: volume_spacing f32 [3]   (z,y,x)
    // 4: trajectory     f32 [360,3,4]
    // 5: projection_multiplier f32 [1]
    // 6: step_size      f32 [1]   (unused by reference)
    const float* sino     = (const float*)d_in[0];
    const float* vorigin  = (const float*)d_in[2];
    const float* vspacing = (const float*)d_in[3];
    const float* traj     = (const float*)d_in[4];
    const float* pmul     = (const float*)d_in[5];
    float* out = (float*)d_out;

    dim3 grid(VY, VZ, 1);   // blockIdx.x = y, blockIdx.y = z
    dim3 block(256, 1, 1);  // 8 waves; each wave = 16 voxels along X
    hipLaunchKernelGGL(cone_bp_wmma_kernel, grid, block, 0, stream,
                       sino, vorigin, vspacing, traj, pmul, out);
}